// Sp_SpectralConv2D_7739531068087
// MI455X (gfx1250) — compile-verified
//
#include <hip/hip_runtime.h>
#include <math.h>

// ---------------------------------------------------------------------------
// FNO SpectralConv2D for MI455X (gfx1250, wave32, WMMA).
// All pruned DFT stages + channel mixes + residual GEMM are expressed as
// f32 WMMA 16x16x4 GEMMs. Residual path fused into the final inverse stage.
// ---------------------------------------------------------------------------

static constexpr int N_  = 8;
static constexpr int R_  = 256;
static constexpr int C_  = 64;   // IN_DIM == OUT_DIM
static constexpr int M_  = 32;   // modes
static constexpr int K2_ = 64;   // 2*M_ (row-mode count: kh in [0,32) U [224,256))

// ---- workspace layout (float offsets) -------------------------------------
static constexpr size_t OFF_A_COS = 0;                       // [32][256]
static constexpr size_t OFF_A_SIN = OFF_A_COS + 32 * 256;
static constexpr size_t OFF_B_RE  = OFF_A_SIN + 32 * 256;    // [64][256]
static constexpr size_t OFF_B_IM  = OFF_B_RE + 64 * 256;
static constexpr size_t OFF_C_RE  = OFF_B_IM + 64 * 256;     // [256][64]
static constexpr size_t OFF_C_IM  = OFF_C_RE + 256 * 64;
static constexpr size_t OFF_D_COS = OFF_C_IM + 256 * 64;     // [256][32]
static constexpr size_t OFF_D_SIN = OFF_D_COS + 256 * 32;
static constexpr size_t OFF_TW_END = OFF_D_SIN + 256 * 32;

static constexpr size_t SZ_F1  = (size_t)N_ * M_ * R_ * C_;  // 4.19M floats
static constexpr size_t SZ_BLK = (size_t)N_ * M_ * K2_ * C_; // 1.05M floats
static constexpr size_t OFF_F1_RE  = OFF_TW_END;
static constexpr size_t OFF_F1_IM  = OFF_F1_RE + SZ_F1;
static constexpr size_t OFF_BLK_RE = OFF_F1_IM + SZ_F1;
static constexpr size_t OFF_BLK_IM = OFF_BLK_RE + SZ_BLK;
static constexpr size_t OFF_MIX_RE = OFF_BLK_IM + SZ_BLK;
static constexpr size_t OFF_MIX_IM = OFF_MIX_RE + SZ_BLK;
static constexpr size_t OFF_G_RE   = OFF_MIX_IM + SZ_BLK;
static constexpr size_t OFF_G_IM   = OFF_G_RE + SZ_F1;
// total ~21.1M floats = ~84.4 MB of workspace

// ---- WMMA f32 16x16x4 helpers ---------------------------------------------
typedef float v2f __attribute__((ext_vector_type(2)));
typedef float v8f __attribute__((ext_vector_type(8)));

__device__ __forceinline__ v8f wmma_f32(v2f a, v2f b, v8f c) {
  // (neg_a, A, neg_b, B, c_mod, C, reuse_a, reuse_b); f32 NEG[0:1] must be 0
  return __builtin_amdgcn_wmma_f32_16x16x4_f32(false, a, false, b, (short)0, c,
                                               false, false);
}

// A fragment (16x4, row-major source): lanes 0-15 -> K={k0,k0+1},
// lanes 16-31 -> K={k0+2,k0+3}, row = m0 + (lane&15).
__device__ __forceinline__ v2f frag_a(const float* __restrict__ A, int ld,
                                      int m0, int k0, int lane) {
  int row = m0 + (lane & 15);
  int kk  = k0 + ((lane >> 4) << 1);
  v2f r;
  r.x = A[row * ld + kk];
  r.y = A[row * ld + kk + 1];
  return r;
}

// B fragment (4x16, row-major source): col = n0 + (lane&15),
// lanes 0-15 -> rows {k0,k0+1}, lanes 16-31 -> rows {k0+2,k0+3}.
__device__ __forceinline__ v2f frag_b(const float* __restrict__ B, int ld,
                                      int k0, int n0, int lane) {
  int col = n0 + (lane & 15);
  int kk  = k0 + ((lane >> 4) << 1);
  v2f r;
  r.x = B[kk * ld + col];
  r.y = B[(kk + 1) * ld + col];
  return r;
}

// ---- twiddle / DFT-matrix build (phase reduced mod 256 -> exact args) ------
__global__ __launch_bounds__(256) void twiddles(float* __restrict__ tw) {
  int t = blockIdx.x * 256 + threadIdx.x;   // launched so t < 16384
  const float STEP  = 6.28318530717958647692f / 256.0f;
  const float INV16 = 0.0625f;              // ortho: 1/sqrt(256) per axis
  if (t < 32 * 256) {                       // stage A: e^{-i 2pi kw w /256}/16
    int kw = t >> 8, w = t & 255;
    int p = (kw * w) & 255;
    float s, c; sincosf(STEP * (float)p, &s, &c);
    tw[OFF_A_COS + t] = c * INV16;
    tw[OFF_A_SIN + t] = -s * INV16;
  }
  if (t < 64 * 256) {                       // stage B: e^{-i 2pi kh h /256}/16
    int kidx = t >> 8, h = t & 255;
    int kh = (kidx < M_) ? kidx : kidx + (R_ - 2 * M_);  // 224..255
    int p = (kh * h) & 255;
    float s, c; sincosf(STEP * (float)p, &s, &c);
    tw[OFF_B_RE + t] = c * INV16;
    tw[OFF_B_IM + t] = -s * INV16;
  }
  if (t < 256 * 64) {                       // stage C: e^{+i 2pi kh h /256}/16
    int h = t >> 6, kidx = t & 63;
    int kh = (kidx < M_) ? kidx : kidx + (R_ - 2 * M_);
    int p = (kh * h) & 255;
    float s, c; sincosf(STEP * (float)p, &s, &c);
    tw[OFF_C_RE + t] = c * INV16;
    tw[OFF_C_IM + t] = s * INV16;
  }
  if (t < 256 * 32) {                       // stage D: hermitian irfft weights
    int w = t >> 5, k = t & 31;
    int p = (k * w) & 255;
    float scale = ((k == 0) ? 1.0f : 2.0f) * INV16;  // Re(c0); double k>=1
    float s, c; sincosf(STEP * (float)p, &s, &c);
    tw[OFF_D_COS + t] = c * scale;
    tw[OFF_D_SIN + t] = s * scale;
  }
}

// ---- stage A: pruned rfft along w as GEMM ---------------------------------
// block = (n,h). F1[kw(32), c(64)] = Acos/Asin(32x256) * x_row(256x64)
__global__ __launch_bounds__(256) void stageA(const float* __restrict__ x,
                                              const float* __restrict__ tw,
                                              float* __restrict__ f1re,
                                              float* __restrict__ f1im) {
  extern __shared__ float smem[];
  float* xrow = smem;                       // [256][64] = 64KB
  int n = blockIdx.x >> 8, h = blockIdx.x & 255;
  const float* xp = x + ((size_t)(n * R_ + h) * R_) * C_;
  for (int i = threadIdx.x; i < R_ * C_ / 4; i += 256)
    ((float4*)xrow)[i] = ((const float4*)xp)[i];
  __syncthreads();

  int wave = threadIdx.x >> 5, lane = threadIdx.x & 31;
  int m0 = (wave >> 2) * 16;                // kw tile (0 or 16)
  int n0 = (wave & 3) * 16;                 // c tile
  const float* Ac = tw + OFF_A_COS;
  const float* As = tw + OFF_A_SIN;
  v8f accC = {}, accS = {};
  for (int k0 = 0; k0 < 256; k0 += 4) {
    v2f b = frag_b(xrow, C_, k0, n0, lane);
    accC = wmma_f32(frag_a(Ac, 256, m0, k0, lane), b, accC);
    accS = wmma_f32(frag_a(As, 256, m0, k0, lane), b, accS);
  }
  int col = n0 + (lane & 15);
  int rb = m0 + ((lane >> 4) << 3);
#pragma unroll
  for (int v = 0; v < 8; ++v) {             // F1[n][kw][h][c]
    size_t idx = ((size_t)(n * M_ + rb + v) * R_ + h) * C_ + col;
    f1re[idx] = accC[v];
    f1im[idx] = accS[v];
  }
}

// ---- stage B: pruned complex FFT along h as GEMM --------------------------
// block = (n,kw). blk[kidx(64), c(64)] = EB(64x256) * F1_slab(256x64), complex
__global__ __launch_bounds__(256) void stageB(const float* __restrict__ tw,
                                              const float* __restrict__ f1re,
                                              const float* __restrict__ f1im,
                                              float* __restrict__ blkre,
                                              float* __restrict__ blkim) {
  extern __shared__ float smem[];
  float* Br = smem;                         // [256][64]
  float* Bi = smem + R_ * C_;               // [256][64]  (128KB total)
  int n = blockIdx.x >> 5, kw = blockIdx.x & 31;
  const float* sr = f1re + ((size_t)(n * M_ + kw) * R_) * C_;
  const float* si = f1im + ((size_t)(n * M_ + kw) * R_) * C_;
  for (int i = threadIdx.x; i < R_ * C_ / 4; i += 256) {
    ((float4*)Br)[i] = ((const float4*)sr)[i];
    ((float4*)Bi)[i] = ((const float4*)si)[i];
  }
  __syncthreads();

  int wave = threadIdx.x >> 5, lane = threadIdx.x & 31;
  const float* Er = tw + OFF_B_RE;
  const float* Ei = tw + OFF_B_IM;
  for (int tile = wave; tile < 16; tile += 8) {
    int m0 = (tile >> 2) * 16;              // kidx tile
    int n0 = (tile & 3) * 16;               // c tile
    v8f crr = {}, cri = {}, cir = {}, cii = {};
    for (int k0 = 0; k0 < 256; k0 += 4) {
      v2f ar = frag_a(Er, 256, m0, k0, lane);
      v2f ai = frag_a(Ei, 256, m0, k0, lane);
      v2f br = frag_b(Br, C_, k0, n0, lane);
      v2f bi = frag_b(Bi, C_, k0, n0, lane);
      crr = wmma_f32(ar, br, crr);
      cii = wmma_f32(ai, bi, cii);
      cri = wmma_f32(ar, bi, cri);
      cir = wmma_f32(ai, br, cir);
    }
    int col = n0 + (lane & 15);
    int rb = m0 + ((lane >> 4) << 3);
#pragma unroll
    for (int v = 0; v < 8; ++v) {           // blk[n][kw][kidx][c]
      size_t idx = ((size_t)(n * M_ + kw) * K2_ + rb + v) * C_ + col;
      blkre[idx] = crr[v] - cii[v];
      blkim[idx] = cri[v] + cir[v];
    }
  }
}

// ---- per-mode complex channel mix via [Re;Im]-stacked 16-row WMMA ---------
// block = (kidx,kw). out[n(8), o(64)] = blk[n,:] * W[:, :] (complex 64x64)
__global__ __launch_bounds__(256) void modeMix(const float* __restrict__ w0,
                                               const float* __restrict__ w1,
                                               const float* __restrict__ blkre,
                                               const float* __restrict__ blkim,
                                               float* __restrict__ mixre,
                                               float* __restrict__ mixim) {
  extern __shared__ float smem[];
  float* As  = smem;                        // [16][64]  stacked [Re;Im]
  float* Wr  = smem + 16 * 64;              // [64][64]
  float* Wi  = Wr + 64 * 64;                // [64][64]
  float* Ssh = Wi + 64 * 64;                // [2][16][64]
  int kidx = blockIdx.x >> 5, kw = blockIdx.x & 31;

  const float* wsrc = (kidx < M_) ? w0 : w1;
  int kh = (kidx < M_) ? kidx : kidx - M_;
  const float* wp = wsrc + ((size_t)(kh * M_ + kw) * C_ * C_) * 2;
  for (int t = threadIdx.x; t < C_ * C_; t += 256) {
    Wr[t] = wp[2 * t];
    Wi[t] = wp[2 * t + 1];
  }
  for (int t = threadIdx.x; t < N_ * C_; t += 256) {
    int nn = t >> 6, i = t & 63;
    size_t src = ((size_t)(nn * M_ + kw) * K2_ + kidx) * C_ + i;
    As[nn * C_ + i] = blkre[src];           // rows 0..7  : Re
    As[(nn + 8) * C_ + i] = blkim[src];     // rows 8..15 : Im
  }
  __syncthreads();

  int wave = threadIdx.x >> 5, lane = threadIdx.x & 31;
  int mat = wave >> 2;                      // 0: *Wr -> S1, 1: *Wi -> S2
  int n0 = (wave & 3) * 16;
  const float* W = mat ? Wi : Wr;
  v8f acc = {};
  for (int k0 = 0; k0 < C_; k0 += 4)
    acc = wmma_f32(frag_a(As, C_, 0, k0, lane), frag_b(W, C_, k0, n0, lane), acc);
  {
    int col = n0 + (lane & 15);
    int rb = (lane >> 4) << 3;
#pragma unroll
    for (int v = 0; v < 8; ++v) Ssh[mat * 1024 + (rb + v) * C_ + col] = acc[v];
  }
  __syncthreads();

  for (int t = threadIdx.x; t < N_ * C_; t += 256) {
    int nn = t >> 6, o = t & 63;
    float s1a = Ssh[0 * 1024 + nn * C_ + o];        // Re*Wr
    float s1b = Ssh[0 * 1024 + (nn + 8) * C_ + o];  // Im*Wr
    float s2a = Ssh[1 * 1024 + nn * C_ + o];        // Re*Wi
    float s2b = Ssh[1 * 1024 + (nn + 8) * C_ + o];  // Im*Wi
    size_t idx = ((size_t)(nn * M_ + kw) * K2_ + kidx) * C_ + o;
    mixre[idx] = s1a - s2b;
    mixim[idx] = s2a + s1b;
  }
}

// ---- stage C: inverse DFT along h (64 nonzero modes) as GEMM --------------
// block = (n,kw). G[h(256), o(64)] = EC(256x64) * mix_slab(64x64), complex
__global__ __launch_bounds__(256) void stageC(const float* __restrict__ tw,
                                              const float* __restrict__ mixre,
                                              const float* __restrict__ mixim,
                                              float* __restrict__ gre,
                                              float* __restrict__ gim) {
  extern __shared__ float smem[];
  float* Br = smem;                         // [64][64]
  float* Bi = smem + K2_ * C_;              // [64][64]
  int n = blockIdx.x >> 5, kw = blockIdx.x & 31;
  const float* sr = mixre + ((size_t)(n * M_ + kw) * K2_) * C_;
  const float* si = mixim + ((size_t)(n * M_ + kw) * K2_) * C_;
  for (int i = threadIdx.x; i < K2_ * C_ / 4; i += 256) {
    ((float4*)Br)[i] = ((const float4*)sr)[i];
    ((float4*)Bi)[i] = ((const float4*)si)[i];
  }
  __syncthreads();

  int wave = threadIdx.x >> 5, lane = threadIdx.x & 31;
  const float* Er = tw + OFF_C_RE;
  const float* Ei = tw + OFF_C_IM;
  for (int tile = wave; tile < 64; tile += 8) {
    int m0 = (tile >> 2) * 16;              // h tile
    int n0 = (tile & 3) * 16;               // o tile
    v8f crr = {}, cri = {}, cir = {}, cii = {};
    for (int k0 = 0; k0 < K2_; k0 += 4) {
      v2f ar = frag_a(Er, K2_, m0, k0, lane);
      v2f ai = frag_a(Ei, K2_, m0, k0, lane);
      v2f br = frag_b(Br, C_, k0, n0, lane);
      v2f bi = frag_b(Bi, C_, k0, n0, lane);
      crr = wmma_f32(ar, br, crr);
      cii = wmma_f32(ai, bi, cii);
      cri = wmma_f32(ar, bi, cri);
      cir = wmma_f32(ai, br, cir);
    }
    int col = n0 + (lane & 15);
    int rb = m0 + ((lane >> 4) << 3);
#pragma unroll
    for (int v = 0; v < 8; ++v) {           // G[n][h][kw][o]
      size_t idx = ((size_t)(n * R_ + rb + v) * M_ + kw) * C_ + col;
      gre[idx] = crr[v] - cii[v];
      gim[idx] = cri[v] + cir[v];
    }
  }
}

// ---- stage D: hermitian inverse along w + fused residual GEMM + SiLU ------
// block = (n,h). acc[w,o] = CD*Gre + x_row*res_w  -  SD*Gim  (+ bias) -> silu
__global__ __launch_bounds__(256) void stageD(const float* __restrict__ x,
                                              const float* __restrict__ resw,
                                              const float* __restrict__ resb,
                                              const float* __restrict__ tw,
                                              const float* __restrict__ gre,
                                              const float* __restrict__ gim,
                                              float* __restrict__ out) {
  extern __shared__ float smem[];
  float* Gr   = smem;                       // [32][64]
  float* Gi   = Gr + M_ * C_;               // [32][64]
  float* xrow = Gi + M_ * C_;               // [256][64]
  float* Wres = xrow + R_ * C_;             // [64][64]
  float* bias = Wres + C_ * C_;             // [64]
  int n = blockIdx.x >> 8, h = blockIdx.x & 255;

  const float* sr = gre + ((size_t)(n * R_ + h) * M_) * C_;
  const float* si = gim + ((size_t)(n * R_ + h) * M_) * C_;
  const float* xp = x + ((size_t)(n * R_ + h) * R_) * C_;
  for (int i = threadIdx.x; i < M_ * C_ / 4; i += 256) {
    ((float4*)Gr)[i] = ((const float4*)sr)[i];
    ((float4*)Gi)[i] = ((const float4*)si)[i];
  }
  for (int i = threadIdx.x; i < R_ * C_ / 4; i += 256)
    ((float4*)xrow)[i] = ((const float4*)xp)[i];
  for (int i = threadIdx.x; i < C_ * C_ / 4; i += 256)
    ((float4*)Wres)[i] = ((const float4*)resw)[i];
  if (threadIdx.x < C_) bias[threadIdx.x] = resb[threadIdx.x];
  __syncthreads();

  int wave = threadIdx.x >> 5, lane = threadIdx.x & 31;
  const float* Dc = tw + OFF_D_COS;         // [256][32]
  const float* Ds = tw + OFF_D_SIN;
  for (int tile = wave; tile < 64; tile += 8) {
    int m0 = (tile >> 2) * 16;              // w tile
    int n0 = (tile & 3) * 16;               // o tile
    v8f accP = {}, accN = {};
    for (int k0 = 0; k0 < M_; k0 += 4) {    // spectral part
      accP = wmma_f32(frag_a(Dc, M_, m0, k0, lane), frag_b(Gr, C_, k0, n0, lane), accP);
      accN = wmma_f32(frag_a(Ds, M_, m0, k0, lane), frag_b(Gi, C_, k0, n0, lane), accN);
    }
    for (int k0 = 0; k0 < C_; k0 += 4)      // fused residual GEMM
      accP = wmma_f32(frag_a(xrow, C_, m0, k0, lane), frag_b(Wres, C_, k0, n0, lane), accP);
    int col = n0 + (lane & 15);
    int rb = m0 + ((lane >> 4) << 3);
    float bb = bias[col];
#pragma unroll
    for (int v = 0; v < 8; ++v) {
      float val = accP[v] - accN[v] + bb;
      float sig = 1.0f / (1.0f + __expf(-val));
      out[((size_t)(n * R_ + h) * R_ + rb + v) * C_ + col] = val * sig;
    }
  }
}

// ---------------------------------------------------------------------------
extern "C" void kernel_launch(void* const* d_in, const int* in_sizes, int n_in,
                              void* d_out, int out_size, void* d_ws,
                              size_t ws_size, hipStream_t stream) {
  const float* x    = (const float*)d_in[0];  // (8,256,256,64)
  const float* w0   = (const float*)d_in[1];  // (32,32,64,64,2)
  const float* w1   = (const float*)d_in[2];  // (32,32,64,64,2)
  const float* resw = (const float*)d_in[3];  // (64,64)
  const float* resb = (const float*)d_in[4];  // (64,)
  float* out = (float*)d_out;
  float* ws  = (float*)d_ws;

  float* tw    = ws;
  float* f1re  = ws + OFF_F1_RE;
  float* f1im  = ws + OFF_F1_IM;
  float* blkre = ws + OFF_BLK_RE;
  float* blkim = ws + OFF_BLK_IM;
  float* mixre = ws + OFF_MIX_RE;
  float* mixim = ws + OFF_MIX_IM;
  float* gre   = ws + OFF_G_RE;
  float* gim   = ws + OFF_G_IM;

  twiddles<<<64, 256, 0, stream>>>(tw);
  stageA<<<N_ * R_, 256, R_ * C_ * sizeof(float), stream>>>(x, tw, f1re, f1im);
  stageB<<<N_ * M_, 256, 2 * R_ * C_ * sizeof(float), stream>>>(tw, f1re, f1im,
                                                                blkre, blkim);
  modeMix<<<K2_ * M_, 256,
            (16 * 64 + 2 * 64 * 64 + 2 * 16 * 64) * sizeof(float), stream>>>(
      w0, w1, blkre, blkim, mixre, mixim);
  stageC<<<N_ * M_, 256, 2 * K2_ * C_ * sizeof(float), stream>>>(tw, mixre,
                                                                 mixim, gre, gim);
  stageD<<<N_ * R_, 256,
           (2 * M_ * C_ + R_ * C_ + C_ * C_ + C_) * sizeof(float), stream>>>(
      x, resw, resb, tw, gre, gim, out);
}